// M04AdaptiveVQ_15195594293439
// MI455X (gfx1250) — compile-verified
//
#include <hip/hip_runtime.h>
#include <hip/hip_bf16.h>

// ---------------------------------------------------------------------------
// Problem constants (match reference)
// ---------------------------------------------------------------------------
static constexpr int Bb   = 8;
static constexpr int Cc   = 1024;   // WavLM feature dim
static constexpr int Tt   = 1500;   // frames
static constexpr int Kk   = 8192;   // centroid bank
static constexpr int Dd   = 256;    // refiner embed dim
static constexpr int Hh   = 512;    // MLP hidden
static constexpr int NQn  = 4;      // RVQ stages
static constexpr int BINSn= 1024;   // bins per stage
static constexpr long BT  = (long)Bb * Tt;          // 12000 rows (divisible by 16)
static constexpr long BTC = BT * Cc;                // 12,288,000
static constexpr float LRELU = 0.01f;
static constexpr float EPSf  = 1e-8f;

typedef __attribute__((ext_vector_type(2))) float v2f;
typedef __attribute__((ext_vector_type(8))) float v8f;

// ---------------------------------------------------------------------------
// f32 WMMA 16x16x4 (exact fp32 matrix op on CDNA5).
// A: 16x4 (lane m=lane&15 holds K = 2*(lane>>4) + {0,1})
// B: 4x16 (lane n=lane&15 holds K = 2*(lane>>4) + {0,1})
// C/D: acc[j] = (row j + 8*(lane>>4), col lane&15)
// ---------------------------------------------------------------------------
static __device__ inline v8f wmma4(v2f a, v2f b, v8f c) {
#if __has_builtin(__builtin_amdgcn_wmma_f32_16x16x4_f32)
  return __builtin_amdgcn_wmma_f32_16x16x4_f32(false, a, false, b, (short)0, c,
                                               false, false);
#else
  const int lane = threadIdx.x & 31;
  const int half = lane >> 4;
  const int col  = lane & 15;
  float a0 = a.x, a1 = a.y, b0 = b.x, b1 = b.y;
#pragma unroll
  for (int j = 0; j < 8; ++j) {
    int r = j + 8 * half;
    float sum = c[j];
    sum += __shfl(a0, r, 32)      * __shfl(b0, col, 32);
    sum += __shfl(a1, r, 32)      * __shfl(b1, col, 32);
    sum += __shfl(a0, 16 + r, 32) * __shfl(b0, 16 + col, 32);
    sum += __shfl(a1, 16 + r, 32) * __shfl(b1, 16 + col, 32);
    c[j] = sum;
  }
  return c;
#endif
}

static __device__ inline v8f vzero8() {
  v8f z = {0.f, 0.f, 0.f, 0.f, 0.f, 0.f, 0.f, 0.f};
  return z;
}

// ---------------------------------------------------------------------------
// Shared WMMA inner pass: NT 16x16 f32 accumulator tiles against a row-major
// (ntab x KD) table.  ONE base pointer; tiles addressed via constant
// IOFFSET immediates (t*16*KD*4 bytes, < 2^23).  Loads are issued in a batch
// (clause-able, staggered waits) before the 16 WMMAs consume them.
// ---------------------------------------------------------------------------
template <int KD, int NT>
static __device__ inline void wmma_kpass(const float* __restrict__ arow,   // LDS
                                         const float* __restrict__ bbase,  // global
                                         v8f (&acc)[NT]) {
#pragma unroll 1
  for (int k0 = 0; k0 < KD; k0 += 16) {
    v2f a[4];
    v2f b[4][NT];
#pragma unroll
    for (int u = 0; u < 4; ++u) a[u] = *(const v2f*)(arow + k0 + 4 * u);
#pragma unroll
    for (int u = 0; u < 4; ++u)
#pragma unroll
      for (int t = 0; t < NT; ++t)
        b[u][t] = *(const v2f*)(bbase + (size_t)t * 16 * KD + k0 + 4 * u);
#pragma unroll
    for (int u = 0; u < 4; ++u)
#pragma unroll
      for (int t = 0; t < NT; ++t)
        acc[t] = wmma4(a[u], b[u][t], acc[t]);
  }
}

// ---------------------------------------------------------------------------
// Row sum-of-squares (one wave32 per row)
// ---------------------------------------------------------------------------
__global__ void rownorm_kernel(const float* __restrict__ m, float* __restrict__ out,
                               int rows, int dim) {
  int gid  = blockIdx.x * blockDim.x + threadIdx.x;
  int row  = gid >> 5;
  int lane = gid & 31;
  if (row >= rows) return;
  const float* p = m + (size_t)row * dim;
  float s = 0.f;
  for (int k = lane; k < dim; k += 32) { float v = p[k]; s += v * v; }
#pragma unroll
  for (int mask = 16; mask >= 1; mask >>= 1) s += __shfl_xor(s, mask, 32);
  if (lane == 0) out[row] = s;
}

// ---------------------------------------------------------------------------
// Generic 2D transpose: in (R x C) -> out (C x R)
// ---------------------------------------------------------------------------
__global__ void transpose2d_kernel(const float* __restrict__ in, float* __restrict__ out,
                                   int R, int Ccols) {
  __shared__ float tile[32][33];
  int r0 = blockIdx.y * 32;
  int c0 = blockIdx.x * 32;
  int x = threadIdx.x, y = threadIdx.y;
#pragma unroll
  for (int i = 0; i < 32; i += 8) {
    int r = r0 + y + i, c = c0 + x;
    if (r < R && c < Ccols) tile[y + i][x] = in[(size_t)r * Ccols + c];
  }
  __syncthreads();
#pragma unroll
  for (int i = 0; i < 32; i += 8) {
    int c = c0 + y + i, r = r0 + x;
    if (r < R && c < Ccols) out[(size_t)c * R + r] = tile[x][y + i];
  }
}

// ---------------------------------------------------------------------------
// (B,C,T) -> (B*T, C) tiled transpose
// ---------------------------------------------------------------------------
__global__ void transpose_in_kernel(const float* __restrict__ in, float* __restrict__ out,
                                    int Cdim, int Tdim) {
  __shared__ float tile[32][33];
  int b  = blockIdx.z;
  int c0 = blockIdx.y * 32;
  int t0 = blockIdx.x * 32;
  int x = threadIdx.x, y = threadIdx.y;
#pragma unroll
  for (int i = 0; i < 32; i += 8) {
    int c = c0 + y + i, t = t0 + x;
    if (t < Tdim) tile[y + i][x] = in[((size_t)b * Cdim + c) * Tdim + t];
  }
  __syncthreads();
#pragma unroll
  for (int i = 0; i < 32; i += 8) {
    int t = t0 + y + i, c = c0 + x;
    if (t < Tdim) out[((size_t)b * Tdim + t) * Cdim + c] = tile[x][y + i];
  }
}

// ---------------------------------------------------------------------------
// Fused nearest-neighbor search:  idx[row] = argmin_n ( ||tab_n||^2 - 2 x.tab_n )
// One block = 16 rows staged in (dynamic) LDS; 8 waves stride the table, each
// wave holding NT 16x16 accumulator tiles.
// ---------------------------------------------------------------------------
template <int KD, int NT>
__global__ void vq_argmin_kernel(const float* __restrict__ rows,
                                 const float* __restrict__ table,
                                 const float* __restrict__ tnorm,
                                 int* __restrict__ out_idx, int ntab) {
  constexpr int LDST = KD + 2;           // pad to dodge bank conflicts
  extern __shared__ float xt[];          // 16 * LDST floats
  __shared__ float redv[8][16];
  __shared__ int   redi[8][16];

  const int  tid     = threadIdx.x;
  const long rowbase = (long)blockIdx.x * 16;

  for (int i = tid; i < 16 * KD; i += 256) {
    int r = i / KD, c = i - r * KD;
    xt[r * LDST + c] = rows[(rowbase + r) * (long)KD + c];
  }
  __syncthreads();

  const int lane = tid & 31, wave = tid >> 5;
  const int half = lane >> 4, mn = lane & 15, ks = half * 2;

  float bestv[8];
  int   besti[8];
#pragma unroll
  for (int j = 0; j < 8; ++j) { bestv[j] = 3.4e38f; besti[j] = 0; }

  const float* arow = xt + mn * LDST + ks;
  constexpr int NSTEP = 128 * NT;        // 8 waves * 16 cols * NT tiles

  for (int n0 = wave * 16 * NT; n0 < ntab; n0 += NSTEP) {
    const float* bbase = table + (size_t)(n0 + mn) * KD + ks;
    if (n0 + NSTEP < ntab)               // pull next chunk toward WGP$
      __builtin_prefetch(bbase + (size_t)NSTEP * KD, 0, 3);
    v8f acc[NT];
#pragma unroll
    for (int t = 0; t < NT; ++t) acc[t] = vzero8();

    wmma_kpass<KD, NT>(arow, bbase, acc);

#pragma unroll
    for (int t = 0; t < NT; ++t) {
      const int col = n0 + 16 * t + mn;
      float tn = tnorm[col];
#pragma unroll
      for (int j = 0; j < 8; ++j) {
        float dist = tn - 2.0f * acc[t][j];    // cols scanned in increasing
        if (dist < bestv[j]) { bestv[j] = dist; besti[j] = col; }  // order
      }
    }
  }

  // reduce across the 16 columns held within each half-wave
#pragma unroll
  for (int j = 0; j < 8; ++j) {
#pragma unroll
    for (int mask = 8; mask >= 1; mask >>= 1) {
      float ov = __shfl_xor(bestv[j], mask, 32);
      int   oi = __shfl_xor(besti[j], mask, 32);
      if (ov < bestv[j] || (ov == bestv[j] && oi < besti[j])) {
        bestv[j] = ov; besti[j] = oi;
      }
    }
  }
  if (mn == 0) {
#pragma unroll
    for (int j = 0; j < 8; ++j) {
      int r = j + 8 * half;
      redv[wave][r] = bestv[j];
      redi[wave][r] = besti[j];
    }
  }
  __syncthreads();
  if (tid < 16) {
    float bv = redv[0][tid];
    int   bi = redi[0][tid];
#pragma unroll
    for (int w = 1; w < 8; ++w) {
      float ov = redv[w][tid];
      int   oi = redi[w][tid];
      if (ov < bv || (ov == bv && oi < bi)) { bv = ov; bi = oi; }
    }
    out_idx[rowbase + tid] = bi;
  }
}

// ---------------------------------------------------------------------------
// GEMM: out(BT x N) = act( A(BT x KD) @ Wt^T + bias ),  Wt is (N x KD)
// pre-transposed so B fragments are contiguous b64 loads (same loop as VQ).
// ACT: 0 = none, 1 = leaky-relu(0.01), 2 = relu. One block = 16 rows.
// ---------------------------------------------------------------------------
template <int KD, int NT, int ACT>
__global__ void gemm_bias_act(const float* __restrict__ A, const float* __restrict__ Wt,
                              const float* __restrict__ bias, float* __restrict__ out,
                              int N) {
  constexpr int LDST = KD + 2;
  extern __shared__ float xt[];
  const int  tid     = threadIdx.x;
  const long rowbase = (long)blockIdx.x * 16;

  for (int i = tid; i < 16 * KD; i += 256) {
    int r = i / KD, c = i - r * KD;
    xt[r * LDST + c] = A[(rowbase + r) * (long)KD + c];
  }
  __syncthreads();

  const int lane = tid & 31, wave = tid >> 5;
  const int half = lane >> 4, mn = lane & 15, ks = half * 2;
  const float* arow = xt + mn * LDST + ks;
  constexpr int NSTEP = 128 * NT;

  for (int n0 = wave * 16 * NT; n0 < N; n0 += NSTEP) {
    const float* bbase = Wt + (size_t)(n0 + mn) * KD + ks;
    v8f acc[NT];
#pragma unroll
    for (int t = 0; t < NT; ++t) acc[t] = vzero8();

    wmma_kpass<KD, NT>(arow, bbase, acc);

#pragma unroll
    for (int t = 0; t < NT; ++t) {
      const int col = n0 + 16 * t + mn;
      const float bv = bias[col];
#pragma unroll
      for (int j = 0; j < 8; ++j) {
        float v = acc[t][j] + bv;
        if (ACT == 1) v = (v > 0.f) ? v : LRELU * v;
        if (ACT == 2) v = fmaxf(v, 0.f);
        out[(size_t)(rowbase + j + 8 * half) * N + col] = v;
      }
    }
  }
}

// ---------------------------------------------------------------------------
// Elementwise glue
// ---------------------------------------------------------------------------
__global__ void gather_sub_kernel(const int* __restrict__ idx,
                                  const float* __restrict__ centroid,
                                  float* __restrict__ lin,
                                  float* __restrict__ x_inout, long total) {
  long i = (long)blockIdx.x * blockDim.x + threadIdx.x;
  if (i >= total) return;
  long row = i >> 10;        // /1024
  int  c   = (int)(i & 1023);
  float v  = centroid[(size_t)idx[row] * Cc + c];
  lin[i] = v;
  x_inout[i] = x_inout[i] - v;   // xT becomes spk_raw
}

__global__ void div_eps_kernel(float* __restrict__ se, const float* __restrict__ nv,
                               long total) {
  long i = (long)blockIdx.x * blockDim.x + threadIdx.x;
  if (i >= total) return;
  se[i] = se[i] / (nv[i] + EPSf);
}

__global__ void rvq_update_kernel(const int* __restrict__ idx,
                                  const float* __restrict__ cb,
                                  float* __restrict__ qsum,
                                  float* __restrict__ resid,
                                  int first, long total) {
  long i = (long)blockIdx.x * blockDim.x + threadIdx.x;
  if (i >= total) return;
  long row = i >> 8;          // /256
  int  c   = (int)(i & 255);
  float q  = cb[(size_t)idx[row] * Dd + c];
  qsum[i]  = first ? q : (qsum[i] + q);
  resid[i] = resid[i] - q;
}

__global__ void mul_kernel(float* __restrict__ a, const float* __restrict__ b,
                           long total) {
  long i = (long)blockIdx.x * blockDim.x + threadIdx.x;
  if (i >= total) return;
  a[i] = a[i] * b[i];
}

// (BT x C) + (BT x C)  -> (B,C,T)
__global__ void add_transpose_kernel(const float* __restrict__ p,
                                     const float* __restrict__ q,
                                     float* __restrict__ out, int Cdim, int Tdim) {
  __shared__ float tile[32][33];
  int b  = blockIdx.z;
  int c0 = blockIdx.y * 32;
  int t0 = blockIdx.x * 32;
  int x = threadIdx.x, y = threadIdx.y;
#pragma unroll
  for (int i = 0; i < 32; i += 8) {
    int t = t0 + y + i, c = c0 + x;
    if (t < Tdim) {
      size_t r = ((size_t)b * Tdim + t) * Cdim + c;
      tile[y + i][x] = p[r] + q[r];
    }
  }
  __syncthreads();
#pragma unroll
  for (int i = 0; i < 32; i += 8) {
    int c = c0 + y + i, t = t0 + x;
    if (t < Tdim) out[((size_t)b * Cdim + c) * Tdim + t] = tile[x][y + i];
  }
}

// ---------------------------------------------------------------------------
// Orchestration
// ---------------------------------------------------------------------------
extern "C" void kernel_launch(void* const* d_in, const int* in_sizes, int n_in,
                              void* d_out, int out_size, void* d_ws, size_t ws_size,
                              hipStream_t stream) {
  (void)in_sizes; (void)n_in; (void)out_size; (void)ws_size;

  const float* feature  = (const float*)d_in[0];
  const float* centroid = (const float*)d_in[1];
  const float* enc_w0 = (const float*)d_in[2];  const float* enc_b0 = (const float*)d_in[3];
  const float* enc_w1 = (const float*)d_in[4];  const float* enc_b1 = (const float*)d_in[5];
  const float* enc_w2 = (const float*)d_in[6];  const float* enc_b2 = (const float*)d_in[7];
  const float* dec_w0 = (const float*)d_in[8];  const float* dec_b0 = (const float*)d_in[9];
  const float* dec_w1 = (const float*)d_in[10]; const float* dec_b1 = (const float*)d_in[11];
  const float* dec_w2 = (const float*)d_in[12]; const float* dec_b2 = (const float*)d_in[13];
  const float* nrm_w0 = (const float*)d_in[14]; const float* nrm_b0 = (const float*)d_in[15];
  const float* nrm_w1 = (const float*)d_in[16]; const float* nrm_b1 = (const float*)d_in[17];
  const float* nrm_w2 = (const float*)d_in[18]; const float* nrm_b2 = (const float*)d_in[19];
  const float* codebooks = (const float*)d_in[20];
  float* out = (float*)d_out;

  // ---- workspace carve (floats, 256B-aligned chunks) ----
  float* wsf = (float*)d_ws;
  size_t off = 0;
  auto alloc = [&](size_t n) -> float* {
    float* p = wsf + off;
    off += (n + 63) & ~(size_t)63;
    return p;
  };
  float* xT   = alloc(BTC);        // (BT,C): feature^T -> spk_raw -> dec output
  float* lin  = alloc(BTC);        // (BT,C): gathered centroids
  float* h0   = alloc(BT * Hh);    // (BT,512)
  float* h1   = alloc(BT * Hh);    // (BT,512)
  float* se   = alloc(BT * Dd);    // (BT,256): spk_enc -> se_norm -> residual
  float* nv   = alloc(BT * Dd);    // (BT,256): norm_vec
  float* qs   = alloc(BT * Dd);    // (BT,256): q_sum -> denorm
  float* cn   = alloc(Kk);         // centroid norms
  float* cbn  = alloc(NQn * BINSn);// codebook norms
  int*   vqidx  = (int*)alloc(BT);
  int*   rvqidx = (int*)alloc(BT);
  // pre-transposed weights (N x KD row-major)
  float* t_enc_w0 = alloc((size_t)Hh * Cc);
  float* t_enc_w1 = alloc((size_t)Hh * Hh);
  float* t_enc_w2 = alloc((size_t)Dd * Hh);
  float* t_nrm_w0 = alloc((size_t)Hh * Cc);
  float* t_nrm_w1 = alloc((size_t)Hh * Hh);
  float* t_nrm_w2 = alloc((size_t)Dd * Hh);
  float* t_dec_w0 = alloc((size_t)Hh * Dd);
  float* t_dec_w1 = alloc((size_t)Hh * Hh);
  float* t_dec_w2 = alloc((size_t)Cc * Hh);

  const int nblk16 = (int)(BT / 16);   // 750
  const dim3 tb256(256);
  const dim3 tblk(32, 8);
  const size_t shm1024 = 16 * (Cc + 2) * sizeof(float);
  const size_t shm512  = 16 * (Hh + 2) * sizeof(float);
  const size_t shm256  = 16 * (Dd + 2) * sizeof(float);

  // 1) table norms
  rownorm_kernel<<<(Kk * 32 + 255) / 256, tb256, 0, stream>>>(centroid, cn, Kk, Cc);
  rownorm_kernel<<<(NQn * BINSn * 32 + 255) / 256, tb256, 0, stream>>>(
      codebooks, cbn, NQn * BINSn, Dd);

  // 1b) pre-transpose all MLP weights:  (KD x N) -> (N x KD)
  auto tpose = [&](const float* w, float* wt, int R, int Ncols) {
    dim3 g((Ncols + 31) / 32, (R + 31) / 32);
    transpose2d_kernel<<<g, tblk, 0, stream>>>(w, wt, R, Ncols);
  };
  tpose(enc_w0, t_enc_w0, Cc, Hh);
  tpose(enc_w1, t_enc_w1, Hh, Hh);
  tpose(enc_w2, t_enc_w2, Hh, Dd);
  tpose(nrm_w0, t_nrm_w0, Cc, Hh);
  tpose(nrm_w1, t_nrm_w1, Hh, Hh);
  tpose(nrm_w2, t_nrm_w2, Hh, Dd);
  tpose(dec_w0, t_dec_w0, Dd, Hh);
  tpose(dec_w1, t_dec_w1, Hh, Hh);
  tpose(dec_w2, t_dec_w2, Hh, Cc);

  // 2) feature (B,C,T) -> xT (BT,C)
  {
    dim3 g((Tt + 31) / 32, Cc / 32, Bb);
    transpose_in_kernel<<<g, tblk, 0, stream>>>(feature, xT, Cc, Tt);
  }

  // 3) linguistic VQ: argmin over 8192 centroids (f32 WMMA distance matmul)
  vq_argmin_kernel<Cc, 4><<<nblk16, tb256, shm1024, stream>>>(
      xT, centroid, cn, vqidx, Kk);

  // 4) gather + residual:  lin = centroid[idx];  xT = xT - lin  (spk_raw)
  gather_sub_kernel<<<(unsigned)((BTC + 255) / 256), tb256, 0, stream>>>(
      vqidx, centroid, lin, xT, BTC);

  // 5) encoder MLP (leaky-relu between layers)
  gemm_bias_act<Cc, 4, 1><<<nblk16, tb256, shm1024, stream>>>(xT, t_enc_w0, enc_b0, h0, Hh);
  gemm_bias_act<Hh, 4, 1><<<nblk16, tb256, shm512,  stream>>>(h0, t_enc_w1, enc_b1, h1, Hh);
  gemm_bias_act<Hh, 2, 0><<<nblk16, tb256, shm512,  stream>>>(h1, t_enc_w2, enc_b2, se, Dd);

  // 6) normalizer MLP (relu after every layer), input = lin
  gemm_bias_act<Cc, 4, 2><<<nblk16, tb256, shm1024, stream>>>(lin, t_nrm_w0, nrm_b0, h0, Hh);
  gemm_bias_act<Hh, 4, 2><<<nblk16, tb256, shm512,  stream>>>(h0, t_nrm_w1, nrm_b1, h1, Hh);
  gemm_bias_act<Hh, 2, 2><<<nblk16, tb256, shm512,  stream>>>(h1, t_nrm_w2, nrm_b2, nv, Dd);

  // 7) se = se / (nv + eps)      (se now = residual for RVQ)
  const long nBTD = BT * Dd;
  div_eps_kernel<<<(unsigned)((nBTD + 255) / 256), tb256, 0, stream>>>(se, nv, nBTD);

  // 8) RVQ stages (forward value of straight-through == q_sum)
  for (int i = 0; i < NQn; ++i) {
    const float* cb = codebooks + (size_t)i * BINSn * Dd;
    vq_argmin_kernel<Dd, 4><<<nblk16, tb256, shm256, stream>>>(
        se, cb, cbn + i * BINSn, rvqidx, BINSn);
    rvq_update_kernel<<<(unsigned)((nBTD + 255) / 256), tb256, 0, stream>>>(
        rvqidx, cb, qs, se, (i == 0) ? 1 : 0, nBTD);
  }

  // 9) denorm: qs *= nv
  mul_kernel<<<(unsigned)((nBTD + 255) / 256), tb256, 0, stream>>>(qs, nv, nBTD);

  // 10) decoder MLP -> dec output into xT buffer (BT,C)
  gemm_bias_act<Dd, 4, 1><<<nblk16, tb256, shm256, stream>>>(qs, t_dec_w0, dec_b0, h0, Hh);
  gemm_bias_act<Hh, 4, 1><<<nblk16, tb256, shm512, stream>>>(h0, t_dec_w1, dec_b1, h1, Hh);
  gemm_bias_act<Hh, 4, 0><<<nblk16, tb256, shm512, stream>>>(h1, t_dec_w2, dec_b2, xT, Cc);

  // 11) out (B,C,T) = lin + dec
  {
    dim3 g((Tt + 31) / 32, Cc / 32, Bb);
    add_transpose_kernel<<<g, tblk, 0, stream>>>(lin, xT, out, Cc, Tt);
  }
}